// VFEFunctional_11209864642971
// MI455X (gfx1250) — compile-verified
//
#include <hip/hip_runtime.h>
#include <math.h>

typedef __attribute__((ext_vector_type(2))) float v2f;
typedef __attribute__((ext_vector_type(8))) float v8f;

#define GB   2
#define NN   512
#define KK   8
#define DG   28
#define EPSF 1e-8f
#define NPTS (GB*NN)

__device__ __forceinline__ int qidx(int a, int b) {           // a < b, so(8) pair index
    return (a * (15 - a)) / 2 + (b - a - 1);
}
__device__ __forceinline__ float embedElem(const float* __restrict__ p, int r, int k) {
    if (r == k) return 0.0f;
    return (r < k) ? p[qidx(r, k)] : -p[qidx(k, r)];
}

// ---------------- prologue: per-point terms ----------------
__global__ void vfe_prep(const float* __restrict__ mu, const float* __restrict__ sigma,
                         float* __restrict__ S, float* __restrict__ contrib) {
    int p = blockIdx.x * blockDim.x + threadIdx.x;
    if (p >= NPTS) return;
    const float* s = sigma + (size_t)p * KK;
    const float* m = mu    + (size_t)p * KK;
    float Ssum = 0.f, sig = 0.f, mu2 = 0.f;
#pragma unroll
    for (int k = 0; k < KK; ++k) {
        Ssum += __logf(s[k] + EPSF);
        sig  += s[k];
        mu2  += m[k] * m[k];
    }
    S[p] = Ssum;
    const float LOG2PIE = 2.8378770664093453f;                 // log(2*pi*e)
    float entropy = 0.5f * (KK * LOG2PIE + Ssum);
    float klprior = 0.5f * (sig + mu2 - (float)KK - Ssum);
    contrib[p] = (-1.0f * entropy + 0.1f * klprior) * (1.0f / NPTS); // ALPHA=1, GAMMA=0.1
}

// ---------------- main pairwise kernel ----------------
// grid: GB*(NN/8) blocks, 128 threads (4 waves). Wave owns 2 i's, loops j in chunks of 16.
__global__ __launch_bounds__(128) void vfe_pairs(const float* __restrict__ mu,
                                                 const float* __restrict__ sigma,
                                                 const float* __restrict__ phi,
                                                 const float* __restrict__ S,
                                                 float* __restrict__ falign) {
    // [wave][tile(2j)][pair(2i x 2j)][col][row]  -> 32 KB
    __shared__ float Msc[4][8][4][8][8];

    const int tid  = threadIdx.x;
    const int w    = tid >> 5;
    const int lane = tid & 31;
    const int b     = blockIdx.x / (NN / 8);
    const int itile = blockIdx.x % (NN / 8);
    const int i0    = itile * 8 + w * 2;

    // ---- A operand: [A_i0 ; A_i1] 16x8, built once per wave (ISA 16x4 f32 layout) ----
    const int row16 = lane & 15;
    const int iLocA = row16 >> 3, rA = row16 & 7;
    const int kb    = (lane >> 4) << 1;                         // lanes>=16 hold K=2,3 of chunk
    const float* phiIA = phi + (size_t)(b * NN + i0 + iLocA) * DG;
    v2f a0, a1;
    a0[0] = embedElem(phiIA, rA, kb + 0);
    a0[1] = embedElem(phiIA, rA, kb + 1);
    a1[0] = embedElem(phiIA, rA, 4 + kb + 0);
    a1[1] = embedElem(phiIA, rA, 4 + kb + 1);

    // ---- lane-per-pair constants (VALU phase): lane = 2 i x 16 j ----
    const int iL   = lane >> 4;
    const int jj   = lane & 15;
    const int tSel = jj >> 1, jl = jj & 1, plSel = (iL << 1) | jl;
    const int iIdx = i0 + iL;
    const float* phiI = phi   + (size_t)(b * NN + iIdx) * DG;
    const float* muI  = mu    + (size_t)(b * NN + iIdx) * KK;
    const float* sigI = sigma + (size_t)(b * NN + iIdx) * KK;
    const float  Si   = S[b * NN + iIdx];
    float sI[8], mI[8];
#pragma unroll
    for (int r = 0; r < 8; ++r) { sI[r] = sigI[r]; mI[r] = muI[r]; }

    float mMax = -3.0e38f, Zz = 0.f, Ww = 0.f;                  // online softmax state

    const int col16 = lane & 15;
    const int jLocB = col16 >> 3, cB = col16 & 7;
    const int rbB   = (lane >> 4) << 1;

    for (int jc = 0; jc < NN; jc += 16) {
        // ===== WMMA phase: 8 tiles of (2i x 2j); M = A_i * A_j for 4 pairs per tile =====
#pragma unroll
        for (int t = 0; t < 8; ++t) {
            const int j0 = jc + 2 * t;
            const float* phiJ = phi + (size_t)(b * NN + j0 + jLocB) * DG;
            v2f b0, b1;
            b0[0] = embedElem(phiJ, rbB + 0, cB);
            b0[1] = embedElem(phiJ, rbB + 1, cB);
            b1[0] = embedElem(phiJ, 4 + rbB + 0, cB);
            b1[1] = embedElem(phiJ, 4 + rbB + 1, cB);
            v8f c = {0.f, 0.f, 0.f, 0.f, 0.f, 0.f, 0.f, 0.f};
            c = __builtin_amdgcn_wmma_f32_16x16x4_f32(false, a0, false, b0, (short)0, c, false, false);
            c = __builtin_amdgcn_wmma_f32_16x16x4_f32(false, a1, false, b1, (short)0, c, false, false);
            // D layout: VGPR v = row v (lanes 0-15) / v+8 (lanes 16-31); col = lane&15
            const int pl = ((lane >> 4) << 1) | (col16 >> 3);
            float* dst = &Msc[w][t][pl][lane & 7][0];
            *(float4*)(dst + 0) = make_float4(c[0], c[1], c[2], c[3]);
            *(float4*)(dst + 4) = make_float4(c[4], c[5], c[6], c[7]);
        }
        asm volatile("s_wait_dscnt 0" ::: "memory");            // pin store->load order (same wave)

        // ===== VALU phase: one pair per lane =====
        const int jIdx = jc + 2 * tSel + jl;
        const float* Mp    = &Msc[w][tSel][plSel][0][0];        // M[r][c] at Mp[c*8+r]
        const float* phiJ2 = phi + (size_t)(b * NN + jIdx) * DG;

        float U[28];                                            // upper triangle of A_ij
        {
            int q = 0;
#pragma unroll
            for (int a = 0; a < 8; ++a)
#pragma unroll
                for (int bb = a + 1; bb < 8; ++bb) {
                    U[q] = (phiI[q] - phiJ2[q]) + 0.5f * (Mp[bb * 8 + a] - Mp[a * 8 + bb]);
                    ++q;
                }
        }
        // exp(A_ij) * mu_j via truncated series action
        const float* muJ = mu + (size_t)(b * NN + jIdx) * KK;
        float wv[8], acc[8];
#pragma unroll
        for (int r = 0; r < 8; ++r) { wv[r] = muJ[r]; acc[r] = wv[r]; }
        const float invk[5] = {1.f, 0.5f, 1.f / 3.f, 0.25f, 0.2f};
#pragma unroll
        for (int k = 0; k < 5; ++k) {
            float y[8] = {0.f, 0.f, 0.f, 0.f, 0.f, 0.f, 0.f, 0.f};
            int q = 0;
#pragma unroll
            for (int a = 0; a < 8; ++a)
#pragma unroll
                for (int bb = a + 1; bb < 8; ++bb) {
                    y[a]  += U[q] * wv[bb];
                    y[bb] -= U[q] * wv[a];
                    ++q;
                }
#pragma unroll
            for (int r = 0; r < 8; ++r) { wv[r] = y[r] * invk[k]; acc[r] += wv[r]; }
        }
        // KL(i||j)
        const float* sigJ = sigma + (size_t)(b * NN + jIdx) * KK;
        const float  Sj   = S[b * NN + jIdx];
        float ssum = 0.f;
#pragma unroll
        for (int r = 0; r < 8; ++r) {
            float vp = sigJ[r] + EPSF;
            float d  = acc[r] - mI[r];
            ssum += (sI[r] + d * d) / vp;
        }
        float kl = 0.5f * (ssum - 8.0f + Sj - Si);
        // online softmax of -kl, accumulate sum(w*kl)
        float x = -kl;
        if (x > mMax) { float sc = __expf(mMax - x); Zz *= sc; Ww *= sc; mMax = x; }
        float e = __expf(x - mMax);
        Zz += e; Ww += kl * e;
    }

    // merge softmax states across the 16 lanes sharing the same i
#pragma unroll
    for (int mask = 8; mask >= 1; mask >>= 1) {
        float mo = __shfl_xor(mMax, mask);
        float Zo = __shfl_xor(Zz,   mask);
        float Wo = __shfl_xor(Ww,   mask);
        float mn = fmaxf(mMax, mo);
        float s1 = __expf(mMax - mn), s2 = __expf(mo - mn);
        Zz = Zz * s1 + Zo * s2;
        Ww = Ww * s1 + Wo * s2;
        mMax = mn;
    }
    if ((lane & 15) == 0) falign[b * NN + iIdx] = Ww / Zz;
}

// ---------------- deterministic final reduction ----------------
__global__ void vfe_final(const float* __restrict__ contrib, const float* __restrict__ falign,
                          float* __restrict__ out) {
    __shared__ float red[256];
    int t = threadIdx.x;
    float s = 0.f;
    for (int p = t; p < NPTS; p += 256)
        s += contrib[p] + (1.0f / NPTS) * falign[p];            // BETA = 1
    red[t] = s;
    __syncthreads();
    for (int off = 128; off > 0; off >>= 1) {
        if (t < off) red[t] += red[t + off];
        __syncthreads();
    }
    if (t == 0) out[0] = red[0];
}

extern "C" void kernel_launch(void* const* d_in, const int* in_sizes, int n_in,
                              void* d_out, int out_size, void* d_ws, size_t ws_size,
                              hipStream_t stream) {
    (void)in_sizes; (void)n_in; (void)out_size; (void)ws_size;
    const float* mu    = (const float*)d_in[0];
    const float* sigma = (const float*)d_in[1];
    const float* phi   = (const float*)d_in[2];
    float* S       = (float*)d_ws;           // [NPTS]
    float* contrib = S + NPTS;               // [NPTS]
    float* falign  = S + 2 * NPTS;           // [NPTS]

    vfe_prep <<<(NPTS + 255) / 256, 256, 0, stream>>>(mu, sigma, S, contrib);
    vfe_pairs<<<GB * (NN / 8), 128, 0, stream>>>(mu, sigma, phi, S, falign);
    vfe_final<<<1, 256, 0, stream>>>(contrib, falign, (float*)d_out);
}